// FeedForward_45002667327943
// MI455X (gfx1250) — compile-verified
//
#include <hip/hip_runtime.h>
#include <math.h>
#include <stdint.h>

// ---- problem constants (match reference) ----
#define DIMC 96
#define HIDC 192
#define C2C  384
#define HH   256
#define WWID 256
#define HW   65536   // 256*256
#define PB   8

typedef float v2f __attribute__((ext_vector_type(2)));
typedef float v8f __attribute__((ext_vector_type(8)));

// ---------------------------------------------------------------------------
// Stage 0: derive per-channel 8x8 circular-conv kernel h_c from fft_filt.
// irfft2(rfft2(x) * F) == circular_conv(x, h) with h = IDFT2 of the
// Hermitian extension of F (F real => h real).
// ---------------------------------------------------------------------------
__global__ void prep_filter_kernel(const float* __restrict__ filt,
                                   float* __restrict__ h) {
  const int c = blockIdx.x;
  const int t = threadIdx.x;          // 0..63
  const int a = t >> 3, b = t & 7;
  const float* F = filt + c * (PB * (PB / 2 + 1)); // 8 x 5 per channel
  float s = 0.f;
  for (int k1 = 0; k1 < 8; ++k1) {
    for (int k2 = 0; k2 < 8; ++k2) {
      float G = (k2 <= 4) ? F[k1 * 5 + k2]
                          : F[((8 - k1) & 7) * 5 + (8 - k2)];
      float ph = 0.78539816339744830961f * (float)(k1 * a + k2 * b); // 2pi/8
      s += G * cosf(ph);
    }
  }
  h[c * 64 + t] = s * (1.f / 64.f);
}

// ---------------------------------------------------------------------------
// Stage 1: channel expand GEMM, t1[o,p] = sum_c w_in[o,c] * x[c,p]
// Wave keeps the whole B panel (96x16 f32 = 48 VGPRs) in registers and
// loops the 24 M tiles -> x is read exactly once from memory.
// ---------------------------------------------------------------------------
__global__ __launch_bounds__(256) void gemm_expand_kernel(
    const float* __restrict__ x,     // [DIMC][HW] (one batch image)
    const float* __restrict__ w_in,  // [C2C][DIMC] row-major
    float* __restrict__ t1)          // [C2C][HW]
{
  const int lane = threadIdx.x & 31;
  const int wv   = blockIdx.x * (blockDim.x >> 5) + (threadIdx.x >> 5);
  const int p0   = wv * 16;
  const int half = lane >> 4;
  const int mn   = lane & 15;

  float b0[24], b1[24];
#pragma unroll
  for (int kk = 0; kk < 24; ++kk) {
    const int c = 4 * kk + 2 * half;
    b0[kk] = x[(size_t)c * HW + p0 + mn];
    b1[kk] = x[(size_t)(c + 1) * HW + p0 + mn];
  }

#pragma unroll 1
  for (int ot = 0; ot < 24; ++ot) {
    v8f acc;
#pragma unroll
    for (int j = 0; j < 8; ++j) acc[j] = 0.f;
    const float* wrow = w_in + (size_t)(ot * 16 + mn) * DIMC + 2 * half;
#pragma unroll
    for (int kk = 0; kk < 24; ++kk) {
      v2f av = *(const v2f*)(wrow + 4 * kk);
      v2f bv; bv.x = b0[kk]; bv.y = b1[kk];
      acc = __builtin_amdgcn_wmma_f32_16x16x4_f32(
          false, av, false, bv, (short)0, acc, false, false);
    }
    float* orow = t1 + (size_t)(ot * 16 + 8 * half) * HW + p0 + mn;
#pragma unroll
    for (int v = 0; v < 8; ++v) orow[(size_t)v * HW] = acc[v];
  }
}

// ---------------------------------------------------------------------------
// Stage 2: blockwise 8x8 circular conv with h_c (FFT-filter equivalent).
// In-place: each workgroup stages its 4 blocks fully in LDS first.
// ---------------------------------------------------------------------------
__global__ __launch_bounds__(256) void fft_apply_kernel(
    float* __restrict__ t, const float* __restrict__ h)
{
  __shared__ float hs[64];
  __shared__ float xs[4][64];
  const int tid = threadIdx.x;
  const int q = tid >> 6, pix = tid & 63;
  const int g = blockIdx.x * 4 + q;       // global 8x8 block id
  const int c = g >> 10;                  // 1024 blocks per channel
  const int blk = g & 1023;
  const int by = blk >> 5, bx = blk & 31;
  const int u = pix >> 3, v = pix & 7;
  const size_t base = (size_t)c * HW + (size_t)(by * 8) * WWID + bx * 8;

  xs[q][pix] = t[base + u * WWID + v];
  if (tid < 64) hs[tid] = h[c * 64 + tid];  // same channel across the 4 blocks
  __syncthreads();

  float s = 0.f;
#pragma unroll
  for (int a = 0; a < 8; ++a)
#pragma unroll
    for (int b = 0; b < 8; ++b)
      s += hs[a * 8 + b] * xs[q][((u - a) & 7) * 8 + ((v - b) & 7)];
  t[base + u * WWID + v] = s;
}

// ---------------------------------------------------------------------------
// Stage 3+4 fused: per output tile (16x16, one channel pair c / c+192):
//   x12_1 = dwconv3(t1[c]),     x12_2 = dwconv3(t1[c+192])   (in LDS)
//   t3[c] = gelu(relu(conv3(x12_1))) * relu(conv5(x12_2))
// Interior tiles fill LDS via CDNA5 async global->LDS copies (ASYNCcnt);
// boundary tiles take a guarded path that zero-pads the halo.
// ---------------------------------------------------------------------------
#define S1W 20   // t1[c]    tile + halo 2
#define S2W 22   // t1[c+192] tile + halo 3
#define X1W 18   // x12_1 tile + halo 1
#define X2W 20   // x12_2 tile + halo 2

__global__ __launch_bounds__(256) void dw_branch_fused_kernel(
    const float* __restrict__ t1, const float* __restrict__ w12,
    const float* __restrict__ w3, const float* __restrict__ w5,
    float* __restrict__ t3)
{
  __shared__ float s1[S1W * S1W];
  __shared__ float s2[S2W * S2W];
  __shared__ float x1[X1W * X1W];
  __shared__ float x2[X2W * X2W];

  const int tid = threadIdx.x;
  const int c    = blockIdx.x >> 8;
  const int tile = blockIdx.x & 255;
  const int oy = (tile >> 4) * 16, ox = (tile & 15) * 16;
  const int ty = tid >> 4, tx = tid & 15;

  const float* g1 = t1 + (size_t)c * HW;
  const float* g2 = t1 + (size_t)(c + HIDC) * HW;

  const bool interior = (oy >= 16) && (oy <= 224 - 16 + 13) && (oy + 18 <= 255) &&
                        (ox >= 16) && (ox + 18 <= 255);
  // (oy in [16,224] => oy-3 >= 13 and oy+18 <= 242; same for ox)

  if (interior) {
    // --- async global -> LDS tile fill (tracked by ASYNCcnt) ---
    for (int i = tid; i < S1W * S1W; i += 256) {
      const int r = i / S1W, cc = i - r * S1W;
      const float* g = g1 + (size_t)(oy - 2 + r) * WWID + (ox - 2 + cc);
      uint32_t lo = (uint32_t)(uintptr_t)(&s1[i]);   // LDS byte offset (flat addr low bits)
      asm volatile("global_load_async_to_lds_b32 %0, %1, off"
                   :: "v"(lo), "v"(g) : "memory");
    }
    for (int i = tid; i < S2W * S2W; i += 256) {
      const int r = i / S2W, cc = i - r * S2W;
      const float* g = g2 + (size_t)(oy - 3 + r) * WWID + (ox - 3 + cc);
      uint32_t lo = (uint32_t)(uintptr_t)(&s2[i]);
      asm volatile("global_load_async_to_lds_b32 %0, %1, off"
                   :: "v"(lo), "v"(g) : "memory");
    }
    asm volatile("s_wait_asynccnt 0x0" ::: "memory");
  } else {
    // --- guarded fill with explicit zero padding ---
    for (int i = tid; i < S1W * S1W; i += 256) {
      const int r = i / S1W, cc = i - r * S1W;
      const int gy = oy - 2 + r, gx = ox - 2 + cc;
      s1[i] = ((unsigned)gy < 256u && (unsigned)gx < 256u)
                  ? g1[(size_t)gy * WWID + gx] : 0.f;
    }
    for (int i = tid; i < S2W * S2W; i += 256) {
      const int r = i / S2W, cc = i - r * S2W;
      const int gy = oy - 3 + r, gx = ox - 3 + cc;
      s2[i] = ((unsigned)gy < 256u && (unsigned)gx < 256u)
                  ? g2[(size_t)gy * WWID + gx] : 0.f;
    }
  }
  __syncthreads();

  // --- first depthwise 3x3 into LDS (zero where x12 position is off-image) ---
  float wc1[9], wc2[9];
#pragma unroll
  for (int k = 0; k < 9; ++k) { wc1[k] = w12[c * 9 + k]; wc2[k] = w12[(c + HIDC) * 9 + k]; }

  for (int i = tid; i < X1W * X1W; i += 256) {
    const int r = i / X1W, cc = i - r * X1W;
    const int gy = oy - 1 + r, gx = ox - 1 + cc;
    float s = 0.f;
    if ((unsigned)gy < 256u && (unsigned)gx < 256u) {
#pragma unroll
      for (int ky = 0; ky < 3; ++ky)
#pragma unroll
        for (int kx = 0; kx < 3; ++kx)
          s += wc1[ky * 3 + kx] * s1[(r + ky) * S1W + (cc + kx)];
    }
    x1[i] = s;
  }
  for (int i = tid; i < X2W * X2W; i += 256) {
    const int r = i / X2W, cc = i - r * X2W;
    const int gy = oy - 2 + r, gx = ox - 2 + cc;
    float s = 0.f;
    if ((unsigned)gy < 256u && (unsigned)gx < 256u) {
#pragma unroll
      for (int ky = 0; ky < 3; ++ky)
#pragma unroll
        for (int kx = 0; kx < 3; ++kx)
          s += wc2[ky * 3 + kx] * s2[(r + ky) * S2W + (cc + kx)];
    }
    x2[i] = s;
  }
  __syncthreads();

  // --- second convs + relu + gelu gate, one output pixel per thread ---
  float s3 = 0.f;
#pragma unroll
  for (int ky = 0; ky < 3; ++ky)
#pragma unroll
    for (int kx = 0; kx < 3; ++kx)
      s3 += w3[c * 9 + ky * 3 + kx] * x1[(ty + ky) * X1W + (tx + kx)];
  float s5 = 0.f;
#pragma unroll
  for (int ky = 0; ky < 5; ++ky)
#pragma unroll
    for (int kx = 0; kx < 5; ++kx)
      s5 += w5[c * 25 + ky * 5 + kx] * x2[(ty + ky) * X2W + (tx + kx)];

  const float a  = fmaxf(s3, 0.f);
  const float b5 = fmaxf(s5, 0.f);
  const float gl = a * 0.5f * (1.f + erff(a * 0.70710678118654752f)); // exact gelu
  t3[(size_t)c * HW + (size_t)(oy + ty) * WWID + (ox + tx)] = gl * b5;
}

// ---------------------------------------------------------------------------
// Stage 5: channel contract GEMM, out[o,p] = sum_c w_out[o,c] * t3[c,p]
// 6 M-tiles accumulated in registers; t3 streamed exactly once.
// ---------------------------------------------------------------------------
__global__ __launch_bounds__(256) void gemm_out_kernel(
    const float* __restrict__ t3,    // [HIDC][HW]
    const float* __restrict__ w_out, // [DIMC][HIDC] row-major
    float* __restrict__ out)         // [DIMC][HW]
{
  const int lane = threadIdx.x & 31;
  const int wv   = blockIdx.x * (blockDim.x >> 5) + (threadIdx.x >> 5);
  const int p0   = wv * 16;
  const int half = lane >> 4;
  const int mn   = lane & 15;

  v8f acc[6];
#pragma unroll
  for (int i = 0; i < 6; ++i)
#pragma unroll
    for (int j = 0; j < 8; ++j) acc[i][j] = 0.f;

#pragma unroll 1
  for (int kk = 0; kk < 48; ++kk) {
    const int c = 4 * kk + 2 * half;
    v2f bv;
    bv.x = t3[(size_t)c * HW + p0 + mn];
    bv.y = t3[(size_t)(c + 1) * HW + p0 + mn];
#pragma unroll
    for (int ot = 0; ot < 6; ++ot) {
      v2f av = *(const v2f*)(w_out + (size_t)(ot * 16 + mn) * HIDC + c);
      acc[ot] = __builtin_amdgcn_wmma_f32_16x16x4_f32(
          false, av, false, bv, (short)0, acc[ot], false, false);
    }
  }
#pragma unroll
  for (int ot = 0; ot < 6; ++ot) {
    float* orow = out + (size_t)(ot * 16 + 8 * half) * HW + p0 + mn;
#pragma unroll
    for (int v = 0; v < 8; ++v) orow[(size_t)v * HW] = acc[ot][v];
  }
}

// ---------------------------------------------------------------------------
extern "C" void kernel_launch(void* const* d_in, const int* in_sizes, int n_in,
                              void* d_out, int out_size, void* d_ws, size_t ws_size,
                              hipStream_t stream) {
  const float* x     = (const float*)d_in[0];  // (4,96,256,256)
  const float* w_in  = (const float*)d_in[1];  // (384,96)
  const float* w_dw  = (const float*)d_in[2];  // (384,1,3,3)
  const float* w_dw3 = (const float*)d_in[3];  // (192,1,3,3)
  const float* w_dw5 = (const float*)d_in[4];  // (192,1,5,5)
  const float* filt  = (const float*)d_in[5];  // (384,1,1,8,5)
  const float* w_out = (const float*)d_in[6];  // (96,192)
  float* out = (float*)d_out;                  // (4,96,256,256)

  char* ws = (char*)d_ws;
  float* h  = (float*)(ws);                                        // 384*64
  float* t1 = (float*)(ws + (size_t)98304);                        // C2*HW
  float* t3 = (float*)(ws + (size_t)98304 + (size_t)C2C * HW * 4); // HID*HW

  prep_filter_kernel<<<C2C, 64, 0, stream>>>(filt, h);

  for (int b = 0; b < 4; ++b) {
    const float* xb = x + (size_t)b * DIMC * HW;
    float* ob = out + (size_t)b * DIMC * HW;
    gemm_expand_kernel<<<HW / 16 / 8, 256, 0, stream>>>(xb, w_in, t1);
    fft_apply_kernel<<<C2C * 1024 / 4, 256, 0, stream>>>(t1, h);
    dw_branch_fused_kernel<<<HIDC * 256, 256, 0, stream>>>(t1, w_dw, w_dw3, w_dw5, t3);
    gemm_out_kernel<<<HW / 16 / 8, 256, 0, stream>>>(t3, w_out, ob);
  }
}